// MoEFeedForward_27633819582946
// MI455X (gfx1250) — compile-verified
//
#include <hip/hip_runtime.h>
#include <hip/hip_bf16.h>

// Problem constants (from reference): B=4,S=1024 -> T=4096, D=512, E=8, F=2048, DF=64
#define TT 4096
#define DD 512
#define EE 8
#define FF 2048
#define DFF 64
#define TM 32     // token tile (rows per expert-GEMM block)
#define KC 64     // F-chunk width for fused two-stage GEMM

typedef __attribute__((ext_vector_type(2))) float v2f;
typedef __attribute__((ext_vector_type(8))) float v8f;

__device__ __forceinline__ v8f wmma_f32_16x16x4(v2f a, v2f b, v8f c) {
  // D = A(16x4) * B(4x16) + C(16x16), all f32, wave32
  return __builtin_amdgcn_wmma_f32_16x16x4_f32(false, a, false, b, (short)0, c, false, false);
}

__device__ __forceinline__ float fast_tanh(float x) {
#if __has_builtin(__builtin_amdgcn_tanhf)
  return __builtin_amdgcn_tanhf(x);   // v_tanh_f32 (CDNA5 TRANS op)
#else
  return tanhf(x);
#endif
}

__device__ __forceinline__ float gelu_tanh(float x) {
  const float c = 0.7978845608028654f; // sqrt(2/pi)
  float u = c * (x + 0.044715f * x * x * x);
  return 0.5f * x * (1.0f + fast_tanh(u));
}

__device__ __forceinline__ void wait_async0() {
#if __has_builtin(__builtin_amdgcn_s_wait_asynccnt)
  __builtin_amdgcn_s_wait_asynccnt(0);
#else
  asm volatile("s_wait_asynccnt 0" ::: "memory");
#endif
}

// ---------------- kernel 1: zero counters ----------------
__global__ void k_init(int* counts) {
  if (threadIdx.x < EE) counts[threadIdx.x] = 0;
}

// ---------------- kernel 2: router (wave per token) ----------------
__global__ __launch_bounds__(256) void k_router(
    const float* __restrict__ x, const float* __restrict__ Wr,
    int* __restrict__ counts, float* __restrict__ probs,
    int* __restrict__ te0, int* __restrict__ te1,
    float* __restrict__ tw0, float* __restrict__ tw1) {
  __shared__ float wrs[DD * EE]; // 16 KB
  for (int i = threadIdx.x; i < DD * EE; i += 256) wrs[i] = Wr[i];
  __syncthreads();

  const int wave = threadIdx.x >> 5;
  const int lane = threadIdx.x & 31;
  const int t = blockIdx.x * 8 + wave;

  float acc[EE];
#pragma unroll
  for (int e = 0; e < EE; ++e) acc[e] = 0.0f;

  const float* xt = x + (size_t)t * DD;
  for (int k = lane; k < DD; k += 32) {
    float xv = xt[k];
#pragma unroll
    for (int e = 0; e < EE; ++e) acc[e] += xv * wrs[k * EE + e];
  }
#pragma unroll
  for (int e = 0; e < EE; ++e) {
#pragma unroll
    for (int off = 16; off > 0; off >>= 1) acc[e] += __shfl_xor(acc[e], off, 32);
  }
  if (lane == 0) {
    float m = acc[0];
#pragma unroll
    for (int e = 1; e < EE; ++e) m = fmaxf(m, acc[e]);
    float p[EE], s = 0.0f;
#pragma unroll
    for (int e = 0; e < EE; ++e) { p[e] = __expf(acc[e] - m); s += p[e]; }
    float inv = 1.0f / s;
#pragma unroll
    for (int e = 0; e < EE; ++e) probs[t * EE + e] = p[e] * inv;
    int e0 = 0;
#pragma unroll
    for (int e = 1; e < EE; ++e) if (p[e] > p[e0]) e0 = e;
    int e1 = (e0 == 0) ? 1 : 0;
#pragma unroll
    for (int e = 0; e < EE; ++e) if (e != e0 && p[e] > p[e1]) e1 = e;
    float w0 = p[e0], w1 = p[e1], wsum = w0 + w1;
    te0[t] = e0; te1[t] = e1;
    tw0[t] = w0 / wsum; tw1[t] = w1 / wsum;
    atomicAdd(&counts[e0], 1);
    atomicAdd(&counts[e1], 1);
  }
}

// ---------------- kernel 3: deterministic aux loss + prefix offsets ----------------
__global__ __launch_bounds__(256) void k_scan_aux(
    const int* __restrict__ counts, const float* __restrict__ probs,
    int* __restrict__ offsets, int* __restrict__ cursor, float* __restrict__ aux_out) {
  __shared__ float red[256];
  __shared__ float esum[EE];
  const int tid = threadIdx.x;
  for (int e = 0; e < EE; ++e) {
    float p = 0.0f;
    for (int t = tid; t < TT; t += 256) p += probs[t * EE + e];
    red[tid] = p;
    __syncthreads();
    for (int s = 128; s > 0; s >>= 1) {
      if (tid < s) red[tid] += red[tid + s];
      __syncthreads();
    }
    if (tid == 0) esum[e] = red[0];
    __syncthreads();
  }
  if (tid == 0) {
    int off = 0;
    float aux = 0.0f;
    for (int e = 0; e < EE; ++e) {
      offsets[e] = off;
      cursor[e] = 0;
      int c = counts[e];
      off += c;
      aux += ((float)c / (float)TT) * (esum[e] / (float)TT);
    }
    *aux_out = (float)EE * aux;
  }
}

// ---------------- kernel 4: build per-expert token lists ----------------
__global__ __launch_bounds__(256) void k_build(
    const int* __restrict__ te0, const int* __restrict__ te1,
    const int* __restrict__ offsets, int* __restrict__ cursor,
    int* __restrict__ tlist, int* __restrict__ slot0, int* __restrict__ slot1) {
  int t = blockIdx.x * 256 + threadIdx.x;
  if (t >= TT) return;
  int e0 = te0[t], e1 = te1[t];
  int p0 = atomicAdd(&cursor[e0], 1);
  int s0 = offsets[e0] + p0;
  tlist[s0] = t; slot0[t] = s0;
  int p1 = atomicAdd(&cursor[e1], 1);
  int s1 = offsets[e1] + p1;
  tlist[s1] = t; slot1[t] = s1;
}

// ---------------- kernel 5: side-input augmentation (block per token) ----------------
__global__ __launch_bounds__(256) void k_augment(
    const float* __restrict__ xf, const float* __restrict__ flow,
    const float* __restrict__ avgq, const float* __restrict__ deltas,
    const float* __restrict__ Wf, const float* __restrict__ Wd,
    float* __restrict__ x0, float* __restrict__ x3, float* __restrict__ x4) {
  const int t = blockIdx.x;
  __shared__ float fs[DFF];
  __shared__ float dl[DFF];
  if (threadIdx.x < DFF) {
    fs[threadIdx.x] = flow[(size_t)t * DFF + threadIdx.x];
    dl[threadIdx.x] = deltas[(size_t)t * DFF + threadIdx.x];
  }
  __syncthreads();
  for (int d = threadIdx.x; d < DD; d += 256) {
    float a0 = 0.0f, a4 = 0.0f;
#pragma unroll 8
    for (int k = 0; k < DFF; ++k) {
      a0 += fs[k] * Wf[k * DD + d];
      a4 += dl[k] * Wd[k * DD + d];
    }
    size_t idx = (size_t)t * DD + d;
    float xv = xf[idx];
    x0[idx] = xv + a0;
    x3[idx] = xv + avgq[idx];
    x4[idx] = xv + a4;
  }
}

// ---------------- kernel 6: fused expert FFN (WMMA f32 + async LDS gather) ----------------
// grid (tiles=128, experts=8), block 256 = 8 waves.
// Each block: async-gather 32 token rows -> LDS; loop F in KC chunks:
//   stage1: hc[32xKC] = gelu(xs[32x512] @ W1[:,chunk] + b1)  (wave -> one 16x16 tile)
//   stage2: oacc[32x512] += hc @ W2[chunk,:]                 (wave -> 32 x 64 slab)
__global__ __launch_bounds__(256) void k_expert_ffn(
    const float* __restrict__ xf, const float* __restrict__ x0,
    const float* __restrict__ x3, const float* __restrict__ x4,
    const float* __restrict__ W1, const float* __restrict__ b1,
    const float* __restrict__ W2, const float* __restrict__ b2,
    const int* __restrict__ counts, const int* __restrict__ offsets,
    const int* __restrict__ tlist, float* __restrict__ out_buf) {
  const int e = blockIdx.y;
  const int cnt = counts[e];
  const int row0 = blockIdx.x * TM;
  if (row0 >= cnt) return;
  const int nvalid = min(TM, cnt - row0);
  const int off = offsets[e];

  __shared__ float xs[TM][DD + 4];   // 66 KB, padded for bank spread
  __shared__ float hcs[TM][KC + 4];  // 8.7 KB

  const float* xsrc = (e == 0) ? x0 : (e == 3) ? x3 : (e == 4) ? x4 : xf;

  // ---- async gather of token rows into LDS (global -> LDS DMA, ASYNCcnt) ----
  {
    const int r = threadIdx.x >> 3;
    const int c0 = (threadIdx.x & 7) * 64;  // 64 floats = 16 x b128 per thread
    if (r < nvalid) {
      const int tok = tlist[off + row0 + r];
      const float* src = xsrc + (size_t)tok * DD + c0;
      unsigned lds_addr = (unsigned)(uintptr_t)(&xs[r][c0]);
#pragma unroll
      for (int i = 0; i < 16; ++i) {
        asm volatile("global_load_async_to_lds_b128 %0, %1, off"
                     :: "v"(lds_addr), "v"(src)
                     : "memory");
        lds_addr += 16;
        src += 4;
      }
    } else {
      float4 z = make_float4(0.f, 0.f, 0.f, 0.f);
      float4* dst = (float4*)&xs[r][c0];
#pragma unroll
      for (int i = 0; i < 16; ++i) dst[i] = z;
    }
    wait_async0();  // this wave's DMAs landed in LDS
  }
  __syncthreads();  // all waves' tiles visible

  const int wave = threadIdx.x >> 5;
  const int lane = threadIdx.x & 31;
  const int hi = lane >> 4;   // selects K pair (A/B) or M+8 (C)
  const int lo = lane & 15;
  const int mb1 = wave >> 2;  // stage1: wave -> tile (mb1,nb1) of hc
  const int nb1 = wave & 3;

  const float* W1e = W1 + (size_t)e * DD * FF;
  const float* W2e = W2 + (size_t)e * FF * DD;
  const float* b1e = b1 + (size_t)e * FF;
  const float* b2e = b2 + (size_t)e * DD;

  v8f oacc[2][4];
#pragma unroll
  for (int mb = 0; mb < 2; ++mb)
#pragma unroll
    for (int nb = 0; nb < 4; ++nb)
      oacc[mb][nb] = (v8f){0, 0, 0, 0, 0, 0, 0, 0};

  const float* xrow1 = &xs[mb1 * 16 + lo][0];

  for (int f0 = 0; f0 < FF; f0 += KC) {
    // prefetch next chunk's weights into cache a chunk ahead
    if (f0 + KC < FF) {
      __builtin_prefetch(W1e + (size_t)(hi * 2) * FF + (f0 + KC) + nb1 * 16 + lo, 0, 3);
      __builtin_prefetch(W2e + (size_t)(f0 + KC + hi * 2) * DD + wave * 64 + lo, 0, 3);
    }

    // ---- stage 1: one 16x16 tile of hc per wave, K = 512, dual accumulators ----
    v8f accA = (v8f){0, 0, 0, 0, 0, 0, 0, 0};
    v8f accB = (v8f){0, 0, 0, 0, 0, 0, 0, 0};
    const int colW = f0 + nb1 * 16 + lo;
    const float* w1col = W1e + colW;
#pragma unroll 2
    for (int k = 0; k < DD; k += 8) {
      const int ak = k + hi * 2;
      v2f a, b;
      a.x = xrow1[ak];
      a.y = xrow1[ak + 1];
      b.x = w1col[(size_t)ak * FF];
      b.y = w1col[(size_t)(ak + 1) * FF];
      accA = wmma_f32_16x16x4(a, b, accA);
      const int ak2 = ak + 4;
      v2f a2, b2v;
      a2.x = xrow1[ak2];
      a2.y = xrow1[ak2 + 1];
      b2v.x = w1col[(size_t)ak2 * FF];
      b2v.y = w1col[(size_t)(ak2 + 1) * FF];
      accB = wmma_f32_16x16x4(a2, b2v, accB);
    }
    v8f acc = accA + accB;
    const float bias = b1e[colW];
    __syncthreads();  // prior stage-2 reads of hcs done
#pragma unroll
    for (int v = 0; v < 8; ++v) {
      hcs[mb1 * 16 + hi * 8 + v][nb1 * 16 + lo] = gelu_tanh(acc[v] + bias);
    }
    __syncthreads();

    // ---- stage 2: oacc[32 x (wave*64..+64)] += hc @ W2chunk ----
#pragma unroll 2
    for (int k = 0; k < KC; k += 4) {
      const int ak = k + hi * 2;
      v2f a0v, a1v;
      a0v.x = hcs[lo][ak];      a0v.y = hcs[lo][ak + 1];
      a1v.x = hcs[16 + lo][ak]; a1v.y = hcs[16 + lo][ak + 1];
      const size_t krow = (size_t)(f0 + ak) * DD;
#pragma unroll
      for (int nb = 0; nb < 4; ++nb) {
        const int col = wave * 64 + nb * 16 + lo;
        v2f b;
        b.x = W2e[krow + col];
        b.y = W2e[krow + DD + col];
        oacc[0][nb] = wmma_f32_16x16x4(a0v, b, oacc[0][nb]);
        oacc[1][nb] = wmma_f32_16x16x4(a1v, b, oacc[1][nb]);
      }
    }
  }

  // ---- epilogue: + b2, store valid rows to out_buf[slot, D] ----
#pragma unroll
  for (int mb = 0; mb < 2; ++mb) {
#pragma unroll
    for (int nb = 0; nb < 4; ++nb) {
      const int col = wave * 64 + nb * 16 + lo;
      const float bv = b2e[col];
#pragma unroll
      for (int v = 0; v < 8; ++v) {
        const int row = mb * 16 + hi * 8 + v;
        if (row < nvalid) {
          out_buf[(size_t)(off + row0 + row) * DD + col] = oacc[mb][nb][v] + bv;
        }
      }
    }
  }
}

// ---------------- kernel 7: weighted combine ----------------
__global__ __launch_bounds__(256) void k_combine(
    const float* __restrict__ out_buf, const int* __restrict__ slot0,
    const int* __restrict__ slot1, const float* __restrict__ tw0,
    const float* __restrict__ tw1, float* __restrict__ final_out) {
  const int idx = blockIdx.x * 256 + threadIdx.x;  // over T*D
  const int t = idx >> 9;
  const int d = idx & (DD - 1);
  final_out[idx] = tw0[t] * out_buf[(size_t)slot0[t] * DD + d] +
                   tw1[t] * out_buf[(size_t)slot1[t] * DD + d];
}

extern "C" void kernel_launch(void* const* d_in, const int* in_sizes, int n_in,
                              void* d_out, int out_size, void* d_ws, size_t ws_size,
                              hipStream_t stream) {
  const float* x     = (const float*)d_in[0];
  const float* flow  = (const float*)d_in[1];
  const float* avgq  = (const float*)d_in[2];
  const float* delt  = (const float*)d_in[3];
  const float* Wr    = (const float*)d_in[4];
  const float* W1    = (const float*)d_in[5];
  const float* b1    = (const float*)d_in[6];
  const float* W2    = (const float*)d_in[7];
  const float* b2    = (const float*)d_in[8];
  const float* Wf    = (const float*)d_in[9];
  const float* Wd    = (const float*)d_in[10];

  float* final_out = (float*)d_out;               // [T*D]
  float* aux_out   = final_out + (size_t)TT * DD; // scalar

  // workspace carve-up
  float* x0      = (float*)d_ws;                    // T*D
  float* x3      = x0 + (size_t)TT * DD;            // T*D
  float* x4      = x3 + (size_t)TT * DD;            // T*D
  float* out_buf = x4 + (size_t)TT * DD;            // 2T*D
  float* probs   = out_buf + (size_t)2 * TT * DD;   // T*E
  float* tw0     = probs + (size_t)TT * EE;         // T
  float* tw1     = tw0 + TT;                        // T
  int* te0     = (int*)(tw1 + TT);                  // T
  int* te1     = te0 + TT;                          // T
  int* slot0   = te1 + TT;                          // T
  int* slot1   = slot0 + TT;                        // T
  int* tlist   = slot1 + TT;                        // 2T
  int* counts  = tlist + 2 * TT;                    // E
  int* offsets = counts + EE;                       // E
  int* cursor  = offsets + EE;                      // E

  k_init<<<1, 32, 0, stream>>>(counts);
  k_router<<<TT / 8, 256, 0, stream>>>(x, Wr, counts, probs, te0, te1, tw0, tw1);
  k_scan_aux<<<1, 256, 0, stream>>>(counts, probs, offsets, cursor, aux_out);
  k_build<<<TT / 256, 256, 0, stream>>>(te0, te1, offsets, cursor, tlist, slot0, slot1);
  k_augment<<<TT, 256, 0, stream>>>(x, flow, avgq, delt, Wf, Wd, x0, x3, x4);
  dim3 ffn_grid(TT / TM, EE);
  k_expert_ffn<<<ffn_grid, 256, 0, stream>>>(x, x0, x3, x4, W1, b1, W2, b2,
                                             counts, offsets, tlist, out_buf);
  k_combine<<<(TT * DD) / 256, 256, 0, stream>>>(out_buf, slot0, slot1, tw0, tw1, final_out);
}